// QuantumWave_2619930050892
// MI455X (gfx1250) — compile-verified
//
#include <hip/hip_runtime.h>
#include <math.h>

#define NQ    14
#define DIM   16384          // 2^14
#define BATCH 256
#define HID   1024
#define PRE   (2*NQ)

typedef __attribute__((ext_vector_type(2))) float v2f;
typedef __attribute__((ext_vector_type(8))) float v8f;

// ---------------------------------------------------------------------------
// Kernel 1: params = x @ w_pre^T + b_pre ; then v[b,q] = (cos t/2, e^{i phi} sin t/2)
// v stored as 4 floats per (b,q): re0, im0, re1, im1
// ---------------------------------------------------------------------------
__global__ __launch_bounds__(32)
void k_params(const float* __restrict__ x, const float* __restrict__ w_pre,
              const float* __restrict__ b_pre, float* __restrict__ v)
{
  __shared__ float par[PRE];
  const int b = blockIdx.x;
  const int t = threadIdx.x;
  if (t < PRE) {
    const float4* x4 = (const float4*)(x + (size_t)b * HID);
    const float4* w4 = (const float4*)(w_pre + (size_t)t * HID);
    float acc = 0.f;
    for (int k = 0; k < HID/4; ++k) {
      float4 a = x4[k], w = w4[k];
      acc += a.x*w.x + a.y*w.y + a.z*w.z + a.w*w.w;
    }
    par[t] = acc + b_pre[t];
  }
  __syncthreads();
  if (t < NQ) {
    float theta = par[2*t];
    float phi   = par[2*t + 1];
    float c = cosf(0.5f * theta);
    float s = sinf(0.5f * theta);
    float* o = v + ((size_t)b * NQ + t) * 4;
    o[0] = c;             o[1] = 0.f;
    o[2] = cosf(phi) * s; o[3] = sinf(phi) * s;
  }
}

// ---------------------------------------------------------------------------
// Kernel 2: make_unitary via modified Gram-Schmidt (equals QR with positive-real
// diag(R), which is exactly Q * diag(R)/|diag(R)| from the reference).
// Output interleaved complex row-major: out[2*(i*n+j)] = Re Q[i][j]
// ---------------------------------------------------------------------------
__device__ void mgs_unitarize(const float* Mre, const float* Mim, int n, float* out)
{
  float Qr[16], Qi[16];
  for (int j = 0; j < n; ++j) {
    float cr[4], ci[4];
    for (int i = 0; i < n; ++i) { cr[i] = Mre[i*n + j]; ci[i] = Mim[i*n + j]; }
    for (int k = 0; k < j; ++k) {
      float rr = 0.f, ri = 0.f;                 // r = q_k^H c
      for (int i = 0; i < n; ++i) {
        rr += Qr[i*n+k]*cr[i] + Qi[i*n+k]*ci[i];
        ri += Qr[i*n+k]*ci[i] - Qi[i*n+k]*cr[i];
      }
      for (int i = 0; i < n; ++i) {             // c -= r * q_k
        cr[i] -= rr*Qr[i*n+k] - ri*Qi[i*n+k];
        ci[i] -= rr*Qi[i*n+k] + ri*Qr[i*n+k];
      }
    }
    float nrm = 0.f;
    for (int i = 0; i < n; ++i) nrm += cr[i]*cr[i] + ci[i]*ci[i];
    float inv = rsqrtf(nrm);
    for (int i = 0; i < n; ++i) { Qr[i*n+j] = cr[i]*inv; Qi[i*n+j] = ci[i]*inv; }
  }
  for (int i = 0; i < n*n; ++i) { out[2*i] = Qr[i]; out[2*i + 1] = Qi[i]; }
}

__global__ __launch_bounds__(32)
void k_unitarize(const float* __restrict__ sq_re, const float* __restrict__ sq_im,
                 const float* __restrict__ ent_re, const float* __restrict__ ent_im,
                 float* __restrict__ squ, float* __restrict__ entu)
{
  const int t = threadIdx.x;
  if (t < NQ) {
    mgs_unitarize(sq_re + t*4, sq_im + t*4, 2, squ + t*8);
  } else if (t < NQ + (NQ - 1)) {
    const int g = t - NQ;
    mgs_unitarize(ent_re + g*16, ent_im + g*16, 4, entu + g*32);
  }
}

// ---------------------------------------------------------------------------
// Kernel 3: full statevector simulation, one batch row per workgroup, state
// held in 128 KB of LDS (CDNA5 WGP has 320 KB). probs written to global.
// Qubit q <-> bit q of the flattened index (reference axis mapping).
// ---------------------------------------------------------------------------
__global__ __launch_bounds__(256)
void k_state(const float* __restrict__ v, const float* __restrict__ squ,
             const float* __restrict__ entu, float* __restrict__ probs)
{
  extern __shared__ float smem[];          // re[DIM], im[DIM] : 131072 bytes
  float* re = smem;
  float* im = smem + DIM;
  __shared__ float g1[NQ * 8];             // 14 x 2x2 complex
  __shared__ float g2[(NQ - 1) * 32];      // 13 x 4x4 complex
  __shared__ float vv[NQ * 4];

  const int tid = threadIdx.x;
  const int b   = blockIdx.x;

  for (int i = tid; i < NQ*8;      i += 256) g1[i] = squ[i];
  for (int i = tid; i < (NQ-1)*32; i += 256) g2[i] = entu[i];
  if (tid < NQ*4) vv[tid] = v[(size_t)b * NQ * 4 + tid];
  __syncthreads();

  // initial product state: amp(i) = prod_q v[q][bit_q(i)]
  for (int i = tid; i < DIM; i += 256) {
    float pr = 1.f, pi = 0.f;
    #pragma unroll
    for (int q = 0; q < NQ; ++q) {
      const int bit = (i >> q) & 1;
      const float ur = vv[q*4 + bit*2], ui = vv[q*4 + bit*2 + 1];
      const float nr = pr*ur - pi*ui;
      pi = pr*ui + pi*ur;
      pr = nr;
    }
    re[i] = pr; im[i] = pi;
  }
  __syncthreads();

  // single-qubit gates: out_j = sum_i U[j][i] a_i  (U row-major)
  for (int q = 0; q < NQ; ++q) {
    const float* u = &g1[q*8];
    const float u00r=u[0],u00i=u[1],u01r=u[2],u01i=u[3];
    const float u10r=u[4],u10i=u[5],u11r=u[6],u11i=u[7];
    const int mlow = (1 << q) - 1;
    for (int p = tid; p < DIM/2; p += 256) {
      const int i0 = ((p & ~mlow) << 1) | (p & mlow);
      const int i1 = i0 | (1 << q);
      const float a0r=re[i0], a0i=im[i0], a1r=re[i1], a1i=im[i1];
      re[i0] = u00r*a0r - u00i*a0i + u01r*a1r - u01i*a1i;
      im[i0] = u00r*a0i + u00i*a0r + u01r*a1i + u01i*a1r;
      re[i1] = u10r*a0r - u10i*a0i + u11r*a1r - u11i*a1i;
      im[i1] = u10r*a0i + u10i*a0r + u11r*a1i + u11i*a1r;
    }
    __syncthreads();
  }

  // two-qubit gates on (g, g+1): index t = 2*bit_{g+1} + bit_g, out_t = sum_s G[t][s] a_s
  for (int g = 0; g < NQ - 1; ++g) {
    float Gr[16], Gi[16];
    #pragma unroll
    for (int e = 0; e < 16; ++e) { Gr[e] = g2[g*32 + 2*e]; Gi[e] = g2[g*32 + 2*e + 1]; }
    const int mlow = (1 << g) - 1;
    for (int p = tid; p < DIM/4; p += 256) {
      const int base = ((p & ~mlow) << 2) | (p & mlow);
      int idx[4];
      idx[0] = base;
      idx[1] = base | (1 << g);
      idx[2] = base | (2 << g);
      idx[3] = base | (3 << g);
      float ar[4], ai[4];
      #pragma unroll
      for (int t = 0; t < 4; ++t) { ar[t] = re[idx[t]]; ai[t] = im[idx[t]]; }
      #pragma unroll
      for (int t = 0; t < 4; ++t) {
        float orr = 0.f, oi = 0.f;
        #pragma unroll
        for (int s = 0; s < 4; ++s) {
          orr += Gr[t*4+s]*ar[s] - Gi[t*4+s]*ai[s];
          oi  += Gr[t*4+s]*ai[s] + Gi[t*4+s]*ar[s];
        }
        re[idx[t]] = orr; im[idx[t]] = oi;
      }
    }
    __syncthreads();
  }

  // probabilities
  for (int i = tid; i < DIM; i += 256) {
    const float r = re[i], m = im[i];
    probs[(size_t)b * DIM + i] = r*r + m*m;
  }
}

// ---------------------------------------------------------------------------
// Kernel 4: out = probs @ w_post^T + b_post via V_WMMA_F32_16X16X4_F32.
// M=256, N=1024, K=16384; both operands K-contiguous -> per-lane float2 frags.
//
// 2x2 register tiling: each wave computes a 32(M) x 32(N) macro-tile (4 WMMA
// tiles). Per K-step of 4 we issue 4 loads {a0,a1,w0,w1} feeding 4 WMMAs
// (1 load : 1 wmma), halving operand traffic per tile vs 1x1 tiling and
// spacing same-accumulator WMMAs 4 apart (hides the WMMA->WMMA RAW hazard).
//
// A 16x4 frag: M = lane&15, K = 2*(lane>>4) + {0,1}  (2 VGPRs = float2)
// B 4x16 frag: N = lane&15, K = 2*(lane>>4) + {0,1}
// C/D v8f:     row = r + 8*(lane>>4), col = lane&15
//
// Block = 128 threads = 4 waves in a 2x2 arrangement -> 64x64 per block.
// Grid (256/64, 1024/64) = (4, 16).
// ---------------------------------------------------------------------------
__global__ __launch_bounds__(128)
void k_gemm(const float* __restrict__ A, const float* __restrict__ W,
            const float* __restrict__ bias, float* __restrict__ out)
{
  const int wave = threadIdx.x >> 5;
  const int lane = threadIdx.x & 31;
  const int l    = lane & 15;
  const int hv   = lane >> 4;
  const int tileM = blockIdx.x * 64 + (wave & 1) * 32;   // this wave's 32-row block
  const int tileN = blockIdx.y * 64 + (wave >> 1) * 32;  // this wave's 32-col block

  const float* aPtr0 = A + (size_t)(tileM + l) * DIM + 2*hv;
  const float* aPtr1 = aPtr0 + (size_t)16 * DIM;
  const float* bPtr0 = W + (size_t)(tileN + l) * DIM + 2*hv;
  const float* bPtr1 = bPtr0 + (size_t)16 * DIM;

  v8f acc00 = {0.f,0.f,0.f,0.f,0.f,0.f,0.f,0.f};
  v8f acc01 = acc00, acc10 = acc00, acc11 = acc00;

  for (int k = 0; k < DIM; k += 16) {
    // WGP-scope prefetch (locality 3) pulls next chunk into near caches
    __builtin_prefetch(aPtr0 + k + 128, 0, 3);
    __builtin_prefetch(aPtr1 + k + 128, 0, 3);
    __builtin_prefetch(bPtr0 + k + 128, 0, 3);
    __builtin_prefetch(bPtr1 + k + 128, 0, 3);
    #pragma unroll
    for (int kk = 0; kk < 16; kk += 4) {
      v2f a0 = *(const v2f*)(aPtr0 + k + kk);
      v2f a1 = *(const v2f*)(aPtr1 + k + kk);
      v2f w0 = *(const v2f*)(bPtr0 + k + kk);
      v2f w1 = *(const v2f*)(bPtr1 + k + kk);
      acc00 = __builtin_amdgcn_wmma_f32_16x16x4_f32(false, a0, false, w0, (short)0, acc00, false, false);
      acc01 = __builtin_amdgcn_wmma_f32_16x16x4_f32(false, a0, false, w1, (short)0, acc01, false, false);
      acc10 = __builtin_amdgcn_wmma_f32_16x16x4_f32(false, a1, false, w0, (short)0, acc10, false, false);
      acc11 = __builtin_amdgcn_wmma_f32_16x16x4_f32(false, a1, false, w1, (short)0, acc11, false, false);
    }
  }

  const int row0 = tileM + hv * 8;
  const int col0 = tileN + l;
  const float bb0 = bias[col0];
  const float bb1 = bias[col0 + 16];
  #pragma unroll
  for (int r = 0; r < 8; ++r) {
    out[(size_t)(row0 + r)      * HID + col0     ] = acc00[r] + bb0;
    out[(size_t)(row0 + r)      * HID + col0 + 16] = acc01[r] + bb1;
    out[(size_t)(row0 + r + 16) * HID + col0     ] = acc10[r] + bb0;
    out[(size_t)(row0 + r + 16) * HID + col0 + 16] = acc11[r] + bb1;
  }
}

// ---------------------------------------------------------------------------
extern "C" void kernel_launch(void* const* d_in, const int* in_sizes, int n_in,
                              void* d_out, int out_size, void* d_ws, size_t ws_size,
                              hipStream_t stream)
{
  const float* x      = (const float*)d_in[0];
  const float* w_pre  = (const float*)d_in[1];
  const float* b_pre  = (const float*)d_in[2];
  const float* w_post = (const float*)d_in[3];
  const float* b_post = (const float*)d_in[4];
  const float* sq_re  = (const float*)d_in[5];
  const float* sq_im  = (const float*)d_in[6];
  const float* ent_re = (const float*)d_in[7];
  const float* ent_im = (const float*)d_in[8];

  float* ws    = (float*)d_ws;
  float* v     = ws;                 // 256*14*4       = 14336 floats
  float* squ   = ws + 14336;         // 14*4*2         =   112 floats
  float* entu  = ws + 14448;         // 13*16*2        =   416 floats
  float* probs = ws + 16384;         // 256*16384      = 4.19M floats (64KB-aligned offset)

  k_params   <<<BATCH, 32, 0, stream>>>(x, w_pre, b_pre, v);
  k_unitarize<<<1,     32, 0, stream>>>(sq_re, sq_im, ent_re, ent_im, squ, entu);
  k_state    <<<BATCH, 256, 2 * DIM * sizeof(float), stream>>>(v, squ, entu, probs);

  dim3 grid(BATCH / 64, HID / 64);
  k_gemm     <<<grid, 128, 0, stream>>>(probs, w_post, b_post, (float*)d_out);
}